// DynamicTransformerLayer_9096740733537
// MI455X (gfx1250) — compile-verified
//
#include <hip/hip_runtime.h>
#include <math.h>

// ---------------------------------------------------------------------------
// Problem constants (match reference)
// ---------------------------------------------------------------------------
#define BB    4
#define NN    1024
#define DIMC  512
#define HH    8
#define DHH   64
#define INNERC 512
#define MLPC  2048
#define EPSC  1e-5f
#define BN    (BB * NN)          // 4096 rows
#define BNH   (BB * NN * HH)     // 32768 (b,n,h) stat entries

typedef __attribute__((ext_vector_type(2))) float v2f;
typedef __attribute__((ext_vector_type(8))) float v8f;
typedef __attribute__((ext_vector_type(4))) int   v4i;

// fp32 WMMA: D(16x16) = A(16x4) * B(4x16) + C     -> v_wmma_f32_16x16x4_f32
__device__ __forceinline__ v8f wmma4(v2f a, v2f b, v8f c) {
  return __builtin_amdgcn_wmma_f32_16x16x4_f32(
      /*neg_a=*/false, a, /*neg_b=*/false, b,
      /*c_mod=*/(short)0, c, /*reuse_a=*/false, /*reuse_b=*/false);
}

// ---------------------------------------------------------------------------
// Global -> LDS 16-byte copy. Prefer the CDNA5 async path (ASYNCcnt, no VGPR
// round-trip); fall back to a plain vector copy if the builtin is missing.
// Builtin signature (per hipcc diagnostic): (v4i AS1* src, v4i AS3* dst,
// imm int offset, imm int cpol).
// ---------------------------------------------------------------------------
#if defined(__gfx1250__) && __has_builtin(__builtin_amdgcn_global_load_async_to_lds_b128)
#define HAVE_ASYNC_LDS 1
#else
#define HAVE_ASYNC_LDS 0
#endif

__device__ __forceinline__ void g2l_b128(float* ldst, const float* gsrc) {
#if HAVE_ASYNC_LDS
  __builtin_amdgcn_global_load_async_to_lds_b128(
      (__attribute__((address_space(1))) v4i*)gsrc,
      (__attribute__((address_space(3))) v4i*)ldst, 0, 0);
#else
  *(float4*)ldst = *(const float4*)gsrc;
#endif
}

__device__ __forceinline__ void async_join() {
#if HAVE_ASYNC_LDS && __has_builtin(__builtin_amdgcn_s_wait_asynccnt)
  __builtin_amdgcn_s_wait_asynccnt(0);
#endif
}

// ---------------------------------------------------------------------------
// Row LayerNorm: one 256-thread block per row of DIMC=512 (2 elems/thread)
// ---------------------------------------------------------------------------
__global__ __launch_bounds__(256)
void ln_rows(const float* __restrict__ X, const float* __restrict__ g,
             const float* __restrict__ b, float* __restrict__ Y) {
  const int row = blockIdx.x;
  const int t = threadIdx.x;
  const float* x = X + (size_t)row * DIMC;
  __shared__ float red[256];
  __shared__ float bcast;
  float v0 = x[t], v1 = x[t + 256];
  red[t] = v0 + v1;
  __syncthreads();
  for (int o = 128; o > 0; o >>= 1) {
    if (t < o) red[t] += red[t + o];
    __syncthreads();
  }
  if (t == 0) bcast = red[0] * (1.0f / DIMC);
  __syncthreads();
  float m = bcast;
  float d0 = v0 - m, d1 = v1 - m;
  __syncthreads();
  red[t] = d0 * d0 + d1 * d1;
  __syncthreads();
  for (int o = 128; o > 0; o >>= 1) {
    if (t < o) red[t] += red[t + o];
    __syncthreads();
  }
  if (t == 0) bcast = rsqrtf(red[0] * (1.0f / DIMC) + EPSC);
  __syncthreads();
  float rstd = bcast;
  float* y = Y + (size_t)row * DIMC;
  y[t]       = d0 * rstd * g[t]       + b[t];
  y[t + 256] = d1 * rstd * g[t + 256] + b[t + 256];
}

// ---------------------------------------------------------------------------
// WMMA GEMM: C[M,Nc] = epi(A[M,K] * W[Nc,K]^T (+bias) (+res))
// Block = 64(M) x 64(N), 4 waves (one 64x16 column slab each).
// K processed in 64-wide chunks staged in LDS (async-to-LDS when available).
// Fragment reads: row stride 68 floats -> lanes land on 4-bank-spaced
// addresses, conflict-free ds_load_2addr_b64 pairs.
// ---------------------------------------------------------------------------
enum { EPI_NONE = 0, EPI_BIAS_RES = 1, EPI_BIAS_GELU = 2 };

template <int EPI>
__global__ __launch_bounds__(128)
void gemm_wmma(const float* __restrict__ A, const float* __restrict__ W,
               const float* __restrict__ bias, const float* __restrict__ res,
               float* __restrict__ C, int M, int Nc, int K) {
  const int lane = threadIdx.x & 31;
  const int wave = threadIdx.x >> 5;
  const int n0b = blockIdx.x * 64;
  const int n0 = n0b + wave * 16;
  const int m0 = blockIdx.y * 64;
  const int r16 = lane & 15;
  const int ko  = (lane >> 4) * 2;

  __shared__ float As[64][68];  // [m][k]  (68-f stride: 16B aligned rows)
  __shared__ float Ws[64][68];  // [n][k]

  v8f acc[4] = {};
  for (int kc = 0; kc < K; kc += 64) {
    __syncthreads();  // protect previous chunk's readers
    for (int t = threadIdx.x; t < 64 * 16; t += 128) {
      int r = t >> 4, c4 = (t & 15) << 2;
      g2l_b128(&As[r][c4], &A[(size_t)(m0 + r) * K + kc + c4]);
      g2l_b128(&Ws[r][c4], &W[(size_t)(n0b + r) * K + kc + c4]);
    }
    async_join();
    __syncthreads();
#pragma unroll
    for (int k = 0; k < 64; k += 4) {
      v2f bfrag;
      bfrag.x = Ws[wave * 16 + r16][k + ko];
      bfrag.y = Ws[wave * 16 + r16][k + ko + 1];
#pragma unroll
      for (int i = 0; i < 4; ++i) {
        v2f afrag;
        afrag.x = As[i * 16 + r16][k + ko];
        afrag.y = As[i * 16 + r16][k + ko + 1];
        acc[i] = wmma4(afrag, bfrag, acc[i]);
      }
    }
  }

  const int nn = n0 + r16;
  const float bv = (EPI == EPI_NONE) ? 0.0f : bias[nn];
#pragma unroll
  for (int i = 0; i < 4; ++i) {
    const int mb = m0 + i * 16 + (lane >> 4) * 8;
#pragma unroll
    for (int j = 0; j < 8; ++j) {
      float v = acc[i][j] + bv;
      size_t idx = (size_t)(mb + j) * Nc + nn;
      if (EPI == EPI_BIAS_GELU) v = 0.5f * v * (1.0f + erff(v * 0.70710678118654752f));
      if (EPI == EPI_BIAS_RES)  v += res[idx];
      C[idx] = v;
    }
  }
}

// ---------------------------------------------------------------------------
// Per-(row, head) stats of F[BN, INNER]: 1/norm, mean, unbiased var over DH
// stat index = row*H + h
// ---------------------------------------------------------------------------
__global__ __launch_bounds__(256)
void head_stats(const float* __restrict__ F, float* __restrict__ invn,
                float* __restrict__ mean, float* __restrict__ var) {
  int id = blockIdx.x * blockDim.x + threadIdx.x;
  if (id >= BNH) return;
  int row = id / HH, h = id % HH;
  const float* p = F + (size_t)row * INNERC + h * DHH;
  float s = 0.f, ss = 0.f;
  for (int d = 0; d < DHH; ++d) {
    float x = p[d];
    s += x;
    ss += x * x;
  }
  float m = s * (1.0f / DHH);
  invn[id] = rsqrtf(ss + 1e-30f);
  mean[id] = m;
  var[id] = (ss - (float)DHH * m * m) * (1.0f / (DHH - 1));
}

// ---------------------------------------------------------------------------
// Column means of fq / fk  -> qg[INNER], kg[INNER]   (grid 2*INNER blocks)
// ---------------------------------------------------------------------------
__global__ __launch_bounds__(256)
void col_mean(const float* __restrict__ Fq, const float* __restrict__ Fk,
              float* __restrict__ qg, float* __restrict__ kg) {
  const int c = blockIdx.x % INNERC;
  const float* F = (blockIdx.x < INNERC) ? Fq : Fk;
  float* out = (blockIdx.x < INNERC) ? qg : kg;
  __shared__ float red[256];
  float s = 0.f;
  for (int r = threadIdx.x; r < BN; r += 256) s += F[(size_t)r * INNERC + c];
  red[threadIdx.x] = s;
  __syncthreads();
  for (int o = 128; o > 0; o >>= 1) {
    if (threadIdx.x < o) red[threadIdx.x] += red[threadIdx.x + o];
    __syncthreads();
  }
  if (threadIdx.x == 0) out[c] = red[0] * (1.0f / BN);
}

// ---------------------------------------------------------------------------
// Tiny per-head gating MLP: feats[2*DH] -> LN+ReLU (64) -> softmax over 3
// One thread per head.
// ---------------------------------------------------------------------------
__global__ __launch_bounds__(32)
void head_mlp(const float* __restrict__ qg, const float* __restrict__ kg,
              const float* __restrict__ W1, const float* __restrict__ b1,
              const float* __restrict__ g,  const float* __restrict__ bb,
              const float* __restrict__ W2, const float* __restrict__ b2,
              float* __restrict__ hw) {
  int h = threadIdx.x;
  if (h >= HH) return;
  float feats[2 * DHH];
  for (int d = 0; d < DHH; ++d) {
    feats[d] = qg[h * DHH + d];
    feats[DHH + d] = kg[h * DHH + d];
  }
  float h1[DHH];
  float s = 0.f;
  for (int i = 0; i < DHH; ++i) {
    float a = b1[i];
    for (int j = 0; j < 2 * DHH; ++j) a += feats[j] * W1[i * 2 * DHH + j];
    h1[i] = a;
    s += a;
  }
  float m = s * (1.0f / DHH);
  float ss = 0.f;
  for (int i = 0; i < DHH; ++i) {
    float d = h1[i] - m;
    ss += d * d;
  }
  float rstd = rsqrtf(ss * (1.0f / DHH) + EPSC);
  for (int i = 0; i < DHH; ++i) {
    float r = (h1[i] - m) * rstd * g[i] + bb[i];
    h1[i] = fmaxf(r, 0.0f);
  }
  float lg[3];
  float mx = -1e30f;
  for (int t = 0; t < 3; ++t) {
    float a = b2[t];
    for (int i = 0; i < DHH; ++i) a += h1[i] * W2[t * DHH + i];
    lg[t] = a;
    mx = fmaxf(mx, a);
  }
  float denom = 0.f;
  for (int t = 0; t < 3; ++t) {
    lg[t] = expf(lg[t] - mx);
    denom += lg[t];
  }
  for (int t = 0; t < 3; ++t) hw[h * 3 + t] = lg[t] / denom;
}

// ---------------------------------------------------------------------------
// Fused attention: block = (h, b, 64 n-rows), 4 waves, each wave 16 rows.
// Loops m in chunks of 16: dots via WMMA (K=64), score transform in D-layout,
// S re-layout through per-wave LDS, then S@V via WMMA into 4 out tiles.
// ---------------------------------------------------------------------------
__global__ __launch_bounds__(128)
void attn_fused(const float* __restrict__ Fq, const float* __restrict__ Fk,
                const float* __restrict__ Fv,
                const float* __restrict__ qinvn, const float* __restrict__ qmean,
                const float* __restrict__ qvar,
                const float* __restrict__ kinvn, const float* __restrict__ kmean,
                const float* __restrict__ kvar,
                const float* __restrict__ hw, float* __restrict__ O) {
  const int h = blockIdx.y % HH;
  const int b = blockIdx.y / HH;
  const int lane = threadIdx.x & 31;
  const int wave = threadIdx.x >> 5;
  const int n0 = blockIdx.x * 64;
  const int nw = n0 + wave * 16;
  const int r16 = lane & 15;
  const int ko = (lane >> 4) * 2;
  const int roff = (lane >> 4) * 8;

  __shared__ float qbuf[64][68];
  __shared__ float kbuf[16][68];
  __shared__ float vbuf[16][68];
  __shared__ float sbuf[4][16][17];

  const float cw = hw[h * 3 + 0];
  const float covw = hw[h * 3 + 1];
  const float vw = hw[h * 3 + 2];
  const float invDH = 1.0f / DHH;

  // stage all 64 q-rows for this block (reused across all m-chunks)
  for (int t = threadIdx.x; t < 64 * 16; t += 128) {
    int r = t >> 4, c4 = (t & 15) << 2;
    g2l_b128(&qbuf[r][c4], &Fq[(size_t)(b * NN + n0 + r) * INNERC + h * DHH + c4]);
  }
  // per-wave row stats (rows j + roff of this wave's 16)
  float rq_in[8], rq_m[8], rq_v[8];
#pragma unroll
  for (int j = 0; j < 8; ++j) {
    int idx = (b * NN + nw + roff + j) * HH + h;
    rq_in[j] = qinvn[idx];
    rq_m[j] = qmean[idx];
    rq_v[j] = qvar[idx];
  }
  v8f acco[4] = {};
  async_join();
  __syncthreads();

  for (int m0 = 0; m0 < NN; m0 += 16) {
    __syncthreads();  // protect kbuf/vbuf from previous iteration's readers
    for (int t = threadIdx.x; t < 16 * 16; t += 128) {
      int r = t >> 4, c4 = (t & 15) << 2;
      size_t src = (size_t)(b * NN + m0 + r) * INNERC + h * DHH + c4;
      g2l_b128(&kbuf[r][c4], &Fk[src]);
      g2l_b128(&vbuf[r][c4], &Fv[src]);
    }
    async_join();
    __syncthreads();

    // dots tile: A = q rows (16 x 64), B = k^T (64 x 16)
    v8f accs = {};
#pragma unroll
    for (int k = 0; k < DHH; k += 4) {
      v2f a, bf;
      a.x = qbuf[wave * 16 + r16][k + ko];
      a.y = qbuf[wave * 16 + r16][k + ko + 1];
      bf.x = kbuf[r16][k + ko];
      bf.y = kbuf[r16][k + ko + 1];
      accs = wmma4(a, bf, accs);
    }

    // score transform (element (M=j+roff, N=r16) per D-layout)
    int kidx = (b * NN + m0 + r16) * HH + h;
    float kin = kinvn[kidx], km = kmean[kidx], kv = kvar[kidx];
#pragma unroll
    for (int j = 0; j < 8; ++j) {
      float d = accs[j];
      float sc = d * (cw * rq_in[j] * kin + covw * invDH)
               + vw * rq_v[j] * kv * invDH
               - covw * rq_m[j] * km;
      sbuf[wave][roff + j][r16] = sc;
    }

    // out += S(16x16) @ V(16x64) : 4 col-tiles, K=16
#pragma unroll
    for (int dt = 0; dt < 4; ++dt) {
#pragma unroll
      for (int k = 0; k < 16; k += 4) {
        v2f a, bf;
        a.x = sbuf[wave][r16][k + ko];
        a.y = sbuf[wave][r16][k + ko + 1];
        bf.x = vbuf[k + ko][dt * 16 + r16];
        bf.y = vbuf[k + ko + 1][dt * 16 + r16];
        acco[dt] = wmma4(a, bf, acco[dt]);
      }
    }
  }

  // write out: O[b,n, h*DH + d] (matches transpose(1,2,0,3).reshape)
#pragma unroll
  for (int dt = 0; dt < 4; ++dt) {
#pragma unroll
    for (int j = 0; j < 8; ++j) {
      int n = nw + roff + j;
      O[(size_t)(b * NN + n) * INNERC + h * DHH + dt * 16 + r16] = acco[dt][j];
    }
  }
}

// ---------------------------------------------------------------------------
// Host-side orchestration
// ---------------------------------------------------------------------------
extern "C" void kernel_launch(void* const* d_in, const int* in_sizes, int n_in,
                              void* d_out, int out_size, void* d_ws, size_t ws_size,
                              hipStream_t stream) {
  const float* q    = (const float*)d_in[0];
  const float* k    = (const float*)d_in[1];
  const float* v    = (const float*)d_in[2];
  const float* Win  = (const float*)d_in[3];
  const float* Wout = (const float*)d_in[4];
  const float* bout = (const float*)d_in[5];
  const float* g1   = (const float*)d_in[6];
  const float* b1n  = (const float*)d_in[7];
  const float* g2   = (const float*)d_in[8];
  const float* b2n  = (const float*)d_in[9];
  const float* Wup  = (const float*)d_in[10];
  const float* bup  = (const float*)d_in[11];
  const float* Wdn  = (const float*)d_in[12];
  const float* bdn  = (const float*)d_in[13];
  const float* wpW1 = (const float*)d_in[14];
  const float* wpb1 = (const float*)d_in[15];
  const float* wpg  = (const float*)d_in[16];
  const float* wpb  = (const float*)d_in[17];
  const float* wpW2 = (const float*)d_in[18];
  const float* wpb2 = (const float*)d_in[19];
  float* out = (float*)d_out;

  // carve workspace
  float* ws = (float*)d_ws;
  size_t off = 0;
  auto carve = [&](size_t n) { float* p = ws + off; off += n; return p; };
  float* xq   = carve((size_t)BN * DIMC);
  float* xk   = carve((size_t)BN * DIMC);
  float* xv   = carve((size_t)BN * DIMC);
  float* fq   = carve((size_t)BN * INNERC);
  float* fk   = carve((size_t)BN * INNERC);
  float* fv   = carve((size_t)BN * INNERC);
  float* qin  = carve(BNH);
  float* qm   = carve(BNH);
  float* qv   = carve(BNH);
  float* kin  = carve(BNH);
  float* km   = carve(BNH);
  float* kv   = carve(BNH);
  float* qg   = carve(INNERC);
  float* kg   = carve(INNERC);
  float* hwts = carve(HH * 3);
  float* O    = carve((size_t)BN * INNERC);
  float* q2   = carve((size_t)BN * DIMC);
  float* x2   = carve((size_t)BN * DIMC);
  float* hmid = carve((size_t)BN * MLPC);
  (void)ws_size; (void)in_sizes; (void)n_in; (void)out_size;

  // 1) pre-attention layernorms
  ln_rows<<<BN, 256, 0, stream>>>(q, g1, b1n, xq);
  ln_rows<<<BN, 256, 0, stream>>>(k, g1, b1n, xk);
  ln_rows<<<BN, 256, 0, stream>>>(v, g1, b1n, xv);

  // 2) projections  f = ln(x) @ Win^T   [4096 x 512]
  dim3 gproj(INNERC / 64, BN / 64);
  gemm_wmma<EPI_NONE><<<gproj, 128, 0, stream>>>(xq, Win, nullptr, nullptr, fq, BN, INNERC, DIMC);
  gemm_wmma<EPI_NONE><<<gproj, 128, 0, stream>>>(xk, Win, nullptr, nullptr, fk, BN, INNERC, DIMC);
  gemm_wmma<EPI_NONE><<<gproj, 128, 0, stream>>>(xv, Win, nullptr, nullptr, fv, BN, INNERC, DIMC);

  // 3) per-(row, head) statistics
  head_stats<<<(BNH + 255) / 256, 256, 0, stream>>>(fq, qin, qm, qv);
  head_stats<<<(BNH + 255) / 256, 256, 0, stream>>>(fk, kin, km, kv);

  // 4) per-head global means + gating MLP -> mixing weights
  col_mean<<<2 * INNERC, 256, 0, stream>>>(fq, fk, qg, kg);
  head_mlp<<<1, 32, 0, stream>>>(qg, kg, wpW1, wpb1, wpg, wpb, wpW2, wpb2, hwts);

  // 5) fused mixed-score attention + scores@V
  dim3 gattn(NN / 64, HH * BB);
  attn_fused<<<gattn, 128, 0, stream>>>(fq, fk, fv, qin, qm, qv, kin, km, kv, hwts, O);

  // 6) output projection + residual:  q2 = q + O @ Wout^T + bout
  dim3 gout(DIMC / 64, BN / 64);
  gemm_wmma<EPI_BIAS_RES><<<gout, 128, 0, stream>>>(O, Wout, bout, q, q2, BN, DIMC, INNERC);

  // 7) second layernorm
  ln_rows<<<BN, 256, 0, stream>>>(q2, g2, b2n, x2);

  // 8) MLP up + exact GELU
  dim3 gup(MLPC / 64, BN / 64);
  gemm_wmma<EPI_BIAS_GELU><<<gup, 128, 0, stream>>>(x2, Wup, bup, nullptr, hmid, BN, MLPC, DIMC);

  // 9) MLP down + residual -> final output
  gemm_wmma<EPI_BIAS_RES><<<gout, 128, 0, stream>>>(hmid, Wdn, bdn, q2, out, BN, DIMC, MLPC);
}